// SCOTHead_58351425683692
// MI455X (gfx1250) — compile-verified
//
#include <hip/hip_runtime.h>
#include <hip/hip_bf16.h>
#include <math.h>

#define NJT 17
#define DH  256           // D == H == 256 in this problem
#define BIG_NEG (-2.0e7f) // -BIG_COST / TAU
#define INV_TAU 20.0f     // 1 / 0.05
#define SELU_SCALE 1.0507009873554804934f
#define SELU_ALPHA 1.6732632423543772848f

typedef __attribute__((ext_vector_type(2))) float v2f;
typedef __attribute__((ext_vector_type(8))) float v8f;

// ---------------------------------------------------------------------------
// GEMM1: h = selu(emb @ W1 + b1).  One wave -> one 16x16 tile, fp32 WMMA.
// ---------------------------------------------------------------------------
__global__ void scot_gemm_h(const float* __restrict__ emb,
                            const float* __restrict__ W1,
                            const float* __restrict__ b1,
                            float* __restrict__ h,
                            int mtiles, int ntiles) {
    int wave = (blockIdx.x * blockDim.x + threadIdx.x) >> 5;
    int lane = threadIdx.x & 31;
    if (wave >= mtiles * ntiles) return;       // wave-uniform: EXEC stays all-1s
    int tileM = wave / ntiles, tileN = wave - tileM * ntiles;
    int m0 = tileM * 16, n0 = tileN * 16;
    int hl = lane >> 4, l = lane & 15;

    v8f acc = {};
    const float* arow = emb + (size_t)(m0 + l) * DH;
    for (int kk = 0; kk < DH; kk += 4) {
        v2f a = *(const v2f*)(arow + kk + 2 * hl);     // A: M=l, K=kk+2*hl..+1
        v2f b;                                          // B: N=l, same K pair
        b.x = W1[(size_t)(kk + 2 * hl) * DH + n0 + l];
        b.y = W1[(size_t)(kk + 2 * hl + 1) * DH + n0 + l];
        acc = __builtin_amdgcn_wmma_f32_16x16x4_f32(false, a, false, b,
                                                    (short)0, acc, false, false);
    }
    float bias = b1[n0 + l];
#pragma unroll
    for (int i = 0; i < 8; ++i) {
        int row = m0 + hl * 8 + i;                      // C layout per ISA 7.12.2
        float x = acc[i] + bias;
        float s = (x > 0.f) ? SELU_SCALE * x
                            : SELU_SCALE * SELU_ALPHA * expm1f(x);
        h[(size_t)row * DH + n0 + l] = s;
    }
}

// ---------------------------------------------------------------------------
// GEMM2: d2[n,j] = max(0, |h_n|^2 + |p_j|^2 - 2 h.p).  One wave -> 16x16 tile.
// ---------------------------------------------------------------------------
__global__ void scot_gemm_d2(const float* __restrict__ h,
                             const float* __restrict__ proto,
                             const float* __restrict__ hn2,
                             const float* __restrict__ pn2,
                             float* __restrict__ d2,
                             int mtiles, int ntiles, int k) {
    int wave = (blockIdx.x * blockDim.x + threadIdx.x) >> 5;
    int lane = threadIdx.x & 31;
    if (wave >= mtiles * ntiles) return;
    int tileM = wave / ntiles, tileN = wave - tileM * ntiles;
    int m0 = tileM * 16, n0 = tileN * 16;
    int hl = lane >> 4, l = lane & 15;

    v8f acc = {};
    const float* arow = h + (size_t)(m0 + l) * DH;
    const float* brow = proto + (size_t)(n0 + l) * DH;  // B = proto^T
    for (int kk = 0; kk < DH; kk += 4) {
        v2f a = *(const v2f*)(arow + kk + 2 * hl);
        v2f b = *(const v2f*)(brow + kk + 2 * hl);
        acc = __builtin_amdgcn_wmma_f32_16x16x4_f32(false, a, false, b,
                                                    (short)0, acc, false, false);
    }
    int j = n0 + l;
    float pj = pn2[j];
#pragma unroll
    for (int i = 0; i < 8; ++i) {
        int row = m0 + hl * 8 + i;
        float v = fmaxf(0.f, hn2[row] + pj - 2.f * acc[i]);
        if (j < k) d2[(size_t)row * k + j] = v;
    }
}

// ---------------------------------------------------------------------------
// Row squared norms: one wave per row of a [nrows x 256] matrix.
// ---------------------------------------------------------------------------
__global__ void scot_rownorm(const float* __restrict__ X, float* __restrict__ out,
                             int nrows) {
    int wave = (blockIdx.x * blockDim.x + threadIdx.x) >> 5;
    int lane = threadIdx.x & 31;
    if (wave >= nrows) return;
    const float* r = X + (size_t)wave * DH;
    float s = 0.f;
    for (int i = lane; i < DH; i += 32) { float v = r[i]; s += v * v; }
#pragma unroll
    for (int m = 16; m; m >>= 1) s += __shfl_xor(s, m, 32);
    if (lane == 0) out[wave] = s;
}

// ---------------------------------------------------------------------------
// Deterministic per-type row lists (thread t scans rows in order -> stable
// reduction order across graph replays; no atomics).
// ---------------------------------------------------------------------------
__global__ void scot_typelists(const int* __restrict__ jt, int* __restrict__ counts,
                               int* __restrict__ lists, int N) {
    int t = threadIdx.x;
    if (t >= NJT) return;
    int c = 0;
    for (int n = 0; n < N; ++n)
        if (jt[n] == t) lists[(size_t)t * N + c++] = n;
    counts[t] = c;
}

__global__ void scot_init_logv(float* __restrict__ log_v, int len) {
    int i = blockIdx.x * blockDim.x + threadIdx.x;
    if (i < len) log_v[i] = 0.f;
}

// ---------------------------------------------------------------------------
// Sinkhorn row pass: log_u[n] = -LSE_j(-20*d2[n,j] + log_v[17j+t_n], log_v[ns]).
// One wave per row; values cached in registers (k <= 512 -> <=16/lane).
// ---------------------------------------------------------------------------
__global__ void scot_row_lse(const float* __restrict__ d2,
                             const float* __restrict__ log_v,
                             const int* __restrict__ jt,
                             float* __restrict__ log_u,
                             int N, int k, int ns) {
    int wave = (blockIdx.x * blockDim.x + threadIdx.x) >> 5;
    int lane = threadIdx.x & 31;
    if (wave >= N) return;
    int t = jt[wave];
    const float* drow = d2 + (size_t)wave * k;

    float vals[16];
    int cnt = 0;
    float m = -INFINITY;
    for (int j = lane; j < k; j += 32) {
        float v = -INV_TAU * drow[j] + log_v[j * NJT + t];
        vals[cnt++] = v;
        m = fmaxf(m, v);
    }
    float tail = log_v[ns];
    if (lane == 0) m = fmaxf(m, tail);
#pragma unroll
    for (int x = 16; x; x >>= 1) m = fmaxf(m, __shfl_xor(m, x, 32));
    float s = 0.f;
    for (int i = 0; i < cnt; ++i) s += expf(vals[i] - m);
    if (lane == 0) s += expf(tail - m);
#pragma unroll
    for (int x = 16; x; x >>= 1) s += __shfl_xor(s, x, 32);
    if (lane == 0) log_u[wave] = -(m + logf(s));
}

// ---------------------------------------------------------------------------
// LSE over all log_u (single block) + dummy column log_v[ns].
// ---------------------------------------------------------------------------
__global__ void scot_lse_all(const float* __restrict__ log_u, float* __restrict__ lse_all,
                             float* __restrict__ log_v, int N, int ns, float log_tail) {
    __shared__ float sm[256];
    int tid = threadIdx.x;
    float m = -INFINITY;
    for (int i = tid; i < N; i += 256) m = fmaxf(m, log_u[i]);
    sm[tid] = m; __syncthreads();
    for (int o = 128; o; o >>= 1) { if (tid < o) sm[tid] = fmaxf(sm[tid], sm[tid + o]); __syncthreads(); }
    m = sm[0]; __syncthreads();
    float s = 0.f;
    for (int i = tid; i < N; i += 256) s += expf(log_u[i] - m);
    sm[tid] = s; __syncthreads();
    for (int o = 128; o; o >>= 1) { if (tid < o) sm[tid] += sm[tid + o]; __syncthreads(); }
    if (tid == 0) {
        float lse = m + logf(sm[0]);
        *lse_all = lse;
        log_v[ns] = log_tail - lse;   // log_tm[last] - LSE(log_u)
    }
}

// ---------------------------------------------------------------------------
// Sinkhorn column pass: one wave per slot s = 17j+t, reducing over type-t rows.
// Fallback to -2e7 + LSE_all replicates JAX exactly when a type is absent.
// ---------------------------------------------------------------------------
__global__ void scot_col_lse(const float* __restrict__ d2,
                             const float* __restrict__ log_u,
                             const int* __restrict__ counts,
                             const int* __restrict__ lists,
                             const float* __restrict__ lse_all,
                             float* __restrict__ log_v,
                             int N, int k, int ns) {
    int wave = (blockIdx.x * blockDim.x + threadIdx.x) >> 5;
    int lane = threadIdx.x & 31;
    if (wave >= ns) return;
    int j = wave / NJT, t = wave - j * NJT;
    int cnt = counts[t];
    const int* list = lists + (size_t)t * N;

    float m = -INFINITY;
    for (int i = lane; i < cnt; i += 32) {
        int n = list[i];
        m = fmaxf(m, -INV_TAU * d2[(size_t)n * k + j] + log_u[n]);
    }
#pragma unroll
    for (int x = 16; x; x >>= 1) m = fmaxf(m, __shfl_xor(m, x, 32));
    float s = 0.f;
    for (int i = lane; i < cnt; i += 32) {
        int n = list[i];
        s += expf(-INV_TAU * d2[(size_t)n * k + j] + log_u[n] - m);
    }
#pragma unroll
    for (int x = 16; x; x >>= 1) s += __shfl_xor(s, x, 32);
    if (lane == 0) {
        float lse = (cnt > 0) ? (m + logf(s)) : (BIG_NEG + *lse_all);
        log_v[wave] = -lse;           // log_tm = 0 for real slots
    }
}

// ---------------------------------------------------------------------------
// Final outputs: logits [N x k] then T [N x 17k].  Per (n,j) thread writes the
// single finite slot + 16 exact zeros (matched underflow of the masked cols).
// ---------------------------------------------------------------------------
__global__ void scot_output(const float* __restrict__ d2,
                            const float* __restrict__ log_u,
                            const float* __restrict__ log_v,
                            const int* __restrict__ jt,
                            float* __restrict__ out,
                            int N, int k) {
    int gid = blockIdx.x * blockDim.x + threadIdx.x;
    if (gid >= N * k) return;
    int n = gid / k, j = gid - n * k;
    int t = jt[n];
    float e = -INV_TAU * d2[(size_t)n * k + j] + log_u[n] + log_v[j * NJT + t];
    float Tv = expf(e);
    out[(size_t)n * k + j] = logf(Tv + 1e-8f);               // logits
    float* Trow = out + (size_t)N * k + (size_t)n * (NJT * k) + (size_t)j * NJT;
#pragma unroll
    for (int tt = 0; tt < NJT; ++tt) Trow[tt] = (tt == t) ? Tv : 0.f;
}

// ---------------------------------------------------------------------------
extern "C" void kernel_launch(void* const* d_in, const int* in_sizes, int n_in,
                              void* d_out, int out_size, void* d_ws, size_t ws_size,
                              hipStream_t stream) {
    const float* emb   = (const float*)d_in[0];
    const float* W1    = (const float*)d_in[1];
    const float* b1    = (const float*)d_in[2];
    const float* proto = (const float*)d_in[3];
    const int*   jt    = (const int*)d_in[4];
    // d_in[5] holds k on device; derive it on host from out_size:
    // out_size = N*k (logits) + N*17k (T) = N*(NJT+1)*k
    const int N  = in_sizes[4];
    const int k  = out_size / (N * (NJT + 1));
    const int ns = NJT * k;
    const float log_tail = logf((float)((N - ns) > 1 ? (N - ns) : 1));

    // Workspace layout (floats)
    float* ws   = (float*)d_ws;
    float* h    = ws;                                   // N*256
    float* d2   = h + (size_t)N * DH;                   // N*k
    float* hn2  = d2 + (size_t)N * k;                   // N
    float* pn2  = hn2 + N;                              // pad(k)
    float* logu = pn2 + (((size_t)k + 31) & ~(size_t)31);
    float* logv = logu + N;                             // ns+1 (padded)
    float* lsea = logv + (((size_t)ns + 1 + 31) & ~(size_t)31);
    int* counts = (int*)(lsea + 32);                    // 17 (padded to 32)
    int* lists  = counts + 32;                          // 17*N

    const int mtiles = N / 16;               // 384
    const int nt_h   = DH / 16;              // 16
    const int nt_d2  = (k + 15) / 16;        // 19

    // 1) h = selu(emb @ W1 + b1)   [fp32 WMMA]
    scot_gemm_h<<<(mtiles * nt_h + 7) / 8, 256, 0, stream>>>(emb, W1, b1, h, mtiles, nt_h);
    // 2) norms
    scot_rownorm<<<(N + 7) / 8, 256, 0, stream>>>(h, hn2, N);
    scot_rownorm<<<(k + 7) / 8, 256, 0, stream>>>(proto, pn2, k);
    // 3) d2 = |h|^2 + |p|^2 - 2 h p^T   [fp32 WMMA]
    scot_gemm_d2<<<(mtiles * nt_d2 + 7) / 8, 256, 0, stream>>>(h, proto, hn2, pn2, d2,
                                                              mtiles, nt_d2, k);
    // 4) deterministic type lists + log_v = 0
    scot_typelists<<<1, 32, 0, stream>>>(jt, counts, lists, N);
    scot_init_logv<<<(ns + 1 + 255) / 256, 256, 0, stream>>>(logv, ns + 1);

    // 5) 30 Sinkhorn iterations (sparse: only type-matched slots are finite)
    for (int it = 0; it < 30; ++it) {
        scot_row_lse<<<(N + 7) / 8, 256, 0, stream>>>(d2, logv, jt, logu, N, k, ns);
        scot_lse_all<<<1, 256, 0, stream>>>(logu, lsea, logv, N, ns, log_tail);
        scot_col_lse<<<(ns + 7) / 8, 256, 0, stream>>>(d2, logu, counts, lists,
                                                       lsea, logv, N, k, ns);
    }

    // 6) T + logits (HBM-bound: ~127 MB write)
    scot_output<<<(N * k + 255) / 256, 256, 0, stream>>>(d2, logu, logv, jt,
                                                         (float*)d_out, N, k);
}